// LightGCN_21354577395745
// MI455X (gfx1250) — compile-verified
//
#include <hip/hip_runtime.h>
#include <hip/hip_bf16.h>

#define EMB_D 128

#if defined(__HIP_DEVICE_COMPILE__) && \
    __has_builtin(__builtin_amdgcn_global_load_async_to_lds_b128) && \
    __has_builtin(__builtin_amdgcn_s_wait_asynccnt)
#define USE_ASYNC 1
#else
#define USE_ASYNC 0
#endif

#if USE_ASYNC
typedef int v4i __attribute__((vector_size(16)));
// 16B-per-lane async DMA: global -> LDS, tracked by ASYNCcnt
__device__ __forceinline__ void async_gather16(const float* g, float* l) {
    __builtin_amdgcn_global_load_async_to_lds_b128(
        (__attribute__((address_space(1))) v4i*)g,
        (__attribute__((address_space(3))) v4i*)l,
        0, 0);
}
#endif

// ---------------------------------------------------------------------------
// 1) concat(emb_user, emb_item) -> cur, and -> acc(d_out); zero the histogram
// ---------------------------------------------------------------------------
__global__ void k_init(const float* __restrict__ eu, const float* __restrict__ ei,
                       float* __restrict__ cur, float* __restrict__ acc,
                       int* __restrict__ counts,
                       long long userElems, long long totalElems, int Nn) {
    long long i = (long long)blockIdx.x * blockDim.x + threadIdx.x;
    if (i < totalElems) {
        float v = (i < userElems) ? eu[i] : ei[i - userElems];
        cur[i] = v;
        acc[i] = v;
    }
    if (i < Nn) counts[i] = 0;
}

// ---------------------------------------------------------------------------
// 2) histogram of edge rows
// ---------------------------------------------------------------------------
__global__ void k_hist(const int* __restrict__ rows, int* __restrict__ counts, int E) {
    int e = blockIdx.x * blockDim.x + threadIdx.x;
    if (e < E) atomicAdd(&counts[rows[e]], 1);
}

// ---------------------------------------------------------------------------
// 3) exclusive scan counts -> row_ptr (and fill cursors), single workgroup
// ---------------------------------------------------------------------------
__global__ void __launch_bounds__(1024)
k_scan(const int* __restrict__ counts, int* __restrict__ row_ptr,
       int* __restrict__ fcur, int Nn) {
    __shared__ int sums[1024];
    const int t   = threadIdx.x;
    const int seg = (Nn + 1023) / 1024;
    const int lo  = t * seg;
    const int hi  = (lo + seg < Nn) ? (lo + seg) : Nn;

    int s = 0;
    for (int i = lo; i < hi; ++i) s += counts[i];
    sums[t] = s;
    __syncthreads();
    // Hillis-Steele inclusive scan over 1024 partials
    for (int off = 1; off < 1024; off <<= 1) {
        int v = (t >= off) ? sums[t - off] : 0;
        __syncthreads();
        sums[t] += v;
        __syncthreads();
    }
    int run = (t == 0) ? 0 : sums[t - 1];
    for (int i = lo; i < hi; ++i) {
        row_ptr[i] = run;
        fcur[i]    = run;
        run += counts[i];
    }
    if (t == 1023) row_ptr[Nn] = sums[1023];
}

// ---------------------------------------------------------------------------
// 4) fill CSR (col, val) via per-row atomic slot allocation
// ---------------------------------------------------------------------------
__global__ void k_fill(const int* __restrict__ rows, const int* __restrict__ cols,
                       const float* __restrict__ vals, int* __restrict__ fcur,
                       int* __restrict__ ccol, float* __restrict__ cval, int E) {
    int e = blockIdx.x * blockDim.x + threadIdx.x;
    if (e < E) {
        int r = rows[e];
        int p = atomicAdd(&fcur[r], 1);
        ccol[p] = cols[e];
        cval[p] = vals[e];
    }
}

// ---------------------------------------------------------------------------
// 5) CSR SpMM, one wave32 per row, lane owns float4 of the 128-wide embedding.
//    Gathered rows are double-buffered through LDS with gfx1250 async copies.
//    Fused layer accumulation: acc = (acc + y) * scale.
// ---------------------------------------------------------------------------
__global__ void __launch_bounds__(256)
k_spmm(const float* __restrict__ x, float* __restrict__ y, float* __restrict__ acc,
       const int* __restrict__ rp, const int* __restrict__ ccol,
       const float* __restrict__ cval, int Nn, float scale) {
    __shared__ __align__(16) float smem[8 * 2 * EMB_D];   // 8 waves * 2 buffers * 128f = 8KB
    const int lane = threadIdx.x & 31;
    const int wid  = threadIdx.x >> 5;
    const int row  = blockIdx.x * 8 + wid;
    if (row >= Nn) return;

    const int beg = rp[row];
    const int end = rp[row + 1];
    const int off = lane * 4;                // 16B per lane, 512B per wave (coalesced)
    float ax = 0.f, ay = 0.f, az = 0.f, aw = 0.f;

#if USE_ASYNC
    float* wbuf = &smem[wid * 2 * EMB_D];
    if (beg < end) {
        int c0 = ccol[beg];
        async_gather16(x + (size_t)c0 * EMB_D + off, wbuf + off);
    }
    int buf = 0;
    for (int k = beg; k < end; ++k) {
        if (k + 1 < end) {  // prefetch next gathered row into the other buffer
            int c1 = ccol[k + 1];
            async_gather16(x + (size_t)c1 * EMB_D + off, wbuf + (buf ^ 1) * EMB_D + off);
            __builtin_amdgcn_s_wait_asynccnt(1);   // wait for the older copy only
        } else {
            __builtin_amdgcn_s_wait_asynccnt(0);
        }
        float v = cval[k];
        const float4 xv = *reinterpret_cast<const float4*>(wbuf + buf * EMB_D + off);
        ax = fmaf(v, xv.x, ax);
        ay = fmaf(v, xv.y, ay);
        az = fmaf(v, xv.z, az);
        aw = fmaf(v, xv.w, aw);
        buf ^= 1;
    }
#else
    for (int k = beg; k < end; ++k) {
        if (k + 1 < end) {
            int c1 = ccol[k + 1];
            __builtin_prefetch(x + (size_t)c1 * EMB_D + off, 0, 0);  // global_prefetch_b8
        }
        int c  = ccol[k];
        float v = cval[k];
        const float4 xv = *reinterpret_cast<const float4*>(x + (size_t)c * EMB_D + off);
        ax = fmaf(v, xv.x, ax);
        ay = fmaf(v, xv.y, ay);
        az = fmaf(v, xv.z, az);
        aw = fmaf(v, xv.w, aw);
    }
#endif

    const size_t o = (size_t)row * EMB_D + off;
    float4 r; r.x = ax; r.y = ay; r.z = az; r.w = aw;
    *reinterpret_cast<float4*>(y + o) = r;

    float4 a = *reinterpret_cast<const float4*>(acc + o);
    a.x = (a.x + ax) * scale;
    a.y = (a.y + ay) * scale;
    a.z = (a.z + az) * scale;
    a.w = (a.w + aw) * scale;
    *reinterpret_cast<float4*>(acc + o) = a;
}

// ---------------------------------------------------------------------------
extern "C" void kernel_launch(void* const* d_in, const int* in_sizes, int n_in,
                              void* d_out, int out_size, void* d_ws, size_t ws_size,
                              hipStream_t stream) {
    const float* emb_user = (const float*)d_in[0];
    const float* emb_item = (const float*)d_in[1];
    const int*   edge_row = (const int*)d_in[2];
    const int*   edge_col = (const int*)d_in[3];
    const float* edge_val = (const float*)d_in[4];

    const int nu = in_sizes[0] / EMB_D;
    const int ni = in_sizes[1] / EMB_D;
    const int Nn = nu + ni;
    const int E  = in_sizes[2];
    const long long userElems  = (long long)nu * EMB_D;
    const long long totalElems = (long long)Nn * EMB_D;

    // workspace layout
    float* cur     = (float*)d_ws;                        // Nn*128 f32
    float* nxt     = cur + (size_t)Nn * EMB_D;            // Nn*128 f32
    int*   counts  = (int*)(nxt + (size_t)Nn * EMB_D);    // Nn
    int*   row_ptr = counts + Nn;                         // Nn+1
    int*   fcur    = row_ptr + (Nn + 1);                  // Nn
    int*   csr_col = fcur + Nn;                           // E
    float* csr_val = (float*)(csr_col + E);               // E

    float* acc = (float*)d_out;

    // 1) init: concat embeddings into cur and acc, zero histogram
    {
        long long nthr = totalElems;
        int blocks = (int)((nthr + 255) / 256);
        k_init<<<blocks, 256, 0, stream>>>(emb_user, emb_item, cur, acc, counts,
                                           userElems, totalElems, Nn);
    }
    // 2) histogram rows
    k_hist<<<(E + 255) / 256, 256, 0, stream>>>(edge_row, counts, E);
    // 3) scan -> row_ptr, fill cursors
    k_scan<<<1, 1024, 0, stream>>>(counts, row_ptr, fcur, Nn);
    // 4) CSR fill
    k_fill<<<(E + 255) / 256, 256, 0, stream>>>(edge_row, edge_col, edge_val,
                                                fcur, csr_col, csr_val, E);

    // 5) 3 propagation layers, atomic-free gather SpMM, fused accumulation.
    //    Final layer folds the /4 into the accumulator update.
    const int spmmBlocks = (Nn + 7) / 8;   // 8 rows (waves) per 256-thread block
    k_spmm<<<spmmBlocks, 256, 0, stream>>>(cur, nxt, acc, row_ptr, csr_col, csr_val,
                                           Nn, 1.0f);
    k_spmm<<<spmmBlocks, 256, 0, stream>>>(nxt, cur, acc, row_ptr, csr_col, csr_val,
                                           Nn, 1.0f);
    k_spmm<<<spmmBlocks, 256, 0, stream>>>(cur, nxt, acc, row_ptr, csr_col, csr_val,
                                           Nn, 0.25f);
}